// MultiHeadAttention_1554778161511
// MI455X (gfx1250) — compile-verified
//
#include <hip/hip_runtime.h>
#include <math.h>

// MI455X / gfx1250, wave32.
// Problem constants (fixed by the reference): B=2,H=8,N=64,T=32,D=32.
#define BH     16          // B*H
#define NSER   64          // N
#define DFEAT  1024        // T*D
#define INV_SCALE (1.0f / 32.0f)   // 1/(sqrt(T*D)*TAU) = 1/32

typedef __attribute__((ext_vector_type(2))) float v2f;
typedef __attribute__((ext_vector_type(4))) float v4f;
typedef __attribute__((ext_vector_type(8))) float v8f;

// ---------------------------------------------------------------------------
// Kernel 1: qk[bh][i][j] = (Q[bh,i,:] . K[bh,j,:]) * INV_SCALE
// One wave (32 threads) per 16x16 output tile, 16 tiles per (b,h).
// Uses V_WMMA_F32_16X16X4_F32: A = Q tile (16xK), B = K tile^T (Kx16).
// Fragment layout (ISA 7.12.2, 32-bit A 16x4): lanes 0-15 carry K={0,1} in
// v0/v1, lanes 16-31 carry K={2,3}; B mirrors this with N on the lane index.
// ---------------------------------------------------------------------------
__global__ __launch_bounds__(32)
void qk_wmma_kernel(const float* __restrict__ Q,
                    const float* __restrict__ K,
                    float* __restrict__ qk) {
    const int tile = blockIdx.x;            // 0..15 (4x4 tiles of 64x64)
    const int bh   = blockIdx.y;            // 0..15
    const int tm   = (tile & 3) << 4;       // tile row origin (i)
    const int tn   = (tile >> 2) << 4;      // tile col origin (j)

    const int lane = threadIdx.x;           // 0..31
    const int row  = lane & 15;
    const int half = lane >> 4;             // selects K-pair {0,1} vs {2,3}

    const float* qrow = Q + (size_t)(bh * NSER + tm + row) * DFEAT + half * 2;
    const float* krow = K + (size_t)(bh * NSER + tn + row) * DFEAT + half * 2;

    v8f c = {};
    #pragma unroll 4
    for (int k = 0; k < DFEAT; k += 4) {
        v2f a = *(const v2f*)(qrow + k);    // global_load_b64
        v2f b = *(const v2f*)(krow + k);
        // D = A*B + C  (v_wmma_f32_16x16x4_f32)
        c = __builtin_amdgcn_wmma_f32_16x16x4_f32(
                /*neg_a=*/false, a, /*neg_b=*/false, b,
                /*c_mod=*/(short)0, c, /*reuse_a=*/false, /*reuse_b=*/false);
    }

    // C/D layout: VGPR r holds M=r (lanes 0-15, N=lane) and M=8+r (lanes 16-31).
    const int n     = lane & 15;
    const int mbase = half << 3;
    float* dst = qk + (size_t)bh * NSER * NSER;
    #pragma unroll
    for (int r = 0; r < 8; ++r) {
        const int m = tm + mbase + r;
        dst[m * NSER + tn + n] = c[r] * INV_SCALE;
    }
}

// ---------------------------------------------------------------------------
// Kernel 2: fused mask + softmax + streaming AV contraction.
// One 256-thread block per (b,h,i). V slices V[b,h,j,i,:,:] are 4KB
// contiguous; thread t owns out elements [4t, 4t+3] -> b128 loads, fully
// coalesced, non-temporal (V is 256MB, read exactly once; keep it out of L2).
// ---------------------------------------------------------------------------
__global__ __launch_bounds__(256)
void av_softmax_kernel(const float* __restrict__ qk,
                       const int*   __restrict__ mask,
                       const float* __restrict__ V,
                       float*       __restrict__ out) {
    __shared__ float w[NSER];

    const int tid = threadIdx.x;            // 0..255
    const int i   = blockIdx.x & (NSER - 1);
    const int bh  = blockIdx.x >> 6;

    const size_t rowoff = ((size_t)bh * NSER + i) * NSER;

    // Load masked, pre-scaled logits into LDS.
    if (tid < NSER) {
        const float s = qk[rowoff + tid];
        const int   m = mask[rowoff + tid];
        w[tid] = (m == 0) ? -INFINITY : s;
    }
    __syncthreads();

    // Row max (64 LDS broadcast reads per thread; trivial VALU cost).
    float mx = -INFINITY;
    #pragma unroll 8
    for (int j = 0; j < NSER; ++j) mx = fmaxf(mx, w[j]);
    __syncthreads();                        // all raw reads done before overwrite

    if (tid < NSER) w[tid] = __expf(w[tid] - mx);
    __syncthreads();

    float sum = 0.0f;
    #pragma unroll 8
    for (int j = 0; j < NSER; ++j) sum += w[j];
    const float inv = 1.0f / sum;           // identical in every thread
    __syncthreads();

    if (tid < NSER) w[tid] *= inv;          // normalized weights in LDS
    __syncthreads();

    // Stream V: out[tf] = sum_j w[j] * V[bh, j, i, tf]
    // j-stride in V = NSER*DFEAT floats = 16384 float4s.
    const size_t vbase = (size_t)bh * NSER * NSER * DFEAT
                       + (size_t)i * DFEAT + (size_t)tid * 4;
    const v4f* vp = (const v4f*)(V + vbase);

    v4f acc = {};
    #pragma unroll 8
    for (int j = 0; j < NSER; ++j) {
        const v4f v = __builtin_nontemporal_load(vp + (size_t)j * (NSER * DFEAT / 4));
        const float wj = w[j];
        acc.x += wj * v.x;
        acc.y += wj * v.y;
        acc.z += wj * v.z;
        acc.w += wj * v.w;
    }

    v4f* op = (v4f*)(out + ((size_t)bh * NSER + i) * DFEAT + (size_t)tid * 4);
    __builtin_nontemporal_store(acc, op);
}

// ---------------------------------------------------------------------------
// Launch: inputs in reference order: Q, K, V, mask. Output fp32.
// Workspace: 16*64*64 floats = 256KB for the qk logits.
// ---------------------------------------------------------------------------
extern "C" void kernel_launch(void* const* d_in, const int* in_sizes, int n_in,
                              void* d_out, int out_size, void* d_ws, size_t ws_size,
                              hipStream_t stream) {
    const float* Q    = (const float*)d_in[0];
    const float* K    = (const float*)d_in[1];
    const float* V    = (const float*)d_in[2];
    const int*   mask = (const int*)d_in[3];
    float*       out  = (float*)d_out;
    float*       qk   = (float*)d_ws;      // BH*NSER*NSER floats = 256 KB

    dim3 g1(16, BH);                       // 16 tiles x 16 (b,h)
    qk_wmma_kernel<<<g1, 32, 0, stream>>>(Q, K, qk);

    av_softmax_kernel<<<BH * NSER, 256, 0, stream>>>(qk, mask, V, out);
}